// ComplexChannelAttention_31482110280381
// MI455X (gfx1250) — compile-verified
//
#include <hip/hip_runtime.h>

typedef __attribute__((ext_vector_type(2))) float v2f;
typedef __attribute__((ext_vector_type(4))) float v4f;
typedef __attribute__((ext_vector_type(8))) float v8f;

#define B_   16
#define C_   256
#define R_   64
#define HW_  16384      // 128*128
#define NCH  4096       // B_*C_

// Workspace layout (floats):
//     0: avg_r [4096]
//  4096: avg_i [4096]
//  8192: max_r [4096]
// 12288: max_i [4096]
// 16384: g_r   [4096]   (gate = sigmoid(a+m), [B,C] row-major)
// 20480: g_i   [4096]

// ---------------------------------------------------------------------------
// Kernel 1: adaptive avg+max pool over HxW for both real and imag planes.
// One block per (b,c) channel; 256 threads, each reads 16 float4 (coalesced).
// ---------------------------------------------------------------------------
__global__ __launch_bounds__(256) void pool_kernel(const float* __restrict__ xr,
                                                   const float* __restrict__ xi,
                                                   float* __restrict__ ws) {
  const int ch  = blockIdx.x;              // 0..4095 == b*256 + c
  const int tid = threadIdx.x;
  const v4f* xr4 = (const v4f*)(xr + (size_t)ch * HW_);
  const v4f* xi4 = (const v4f*)(xi + (size_t)ch * HW_);

  float sr = 0.f, si = 0.f;
  float mr = -__builtin_inff(), mi = -__builtin_inff();
#pragma unroll
  for (int t = 0; t < 16; ++t) {
    v4f a = __builtin_nontemporal_load(&xr4[tid + t * 256]);
    v4f b = __builtin_nontemporal_load(&xi4[tid + t * 256]);
    sr += (a.x + a.y) + (a.z + a.w);
    si += (b.x + b.y) + (b.z + b.w);
    mr = fmaxf(mr, fmaxf(fmaxf(a.x, a.y), fmaxf(a.z, a.w)));
    mi = fmaxf(mi, fmaxf(fmaxf(b.x, b.y), fmaxf(b.z, b.w)));
  }

  __shared__ float s0[256], s1[256], s2[256], s3[256];
  s0[tid] = sr; s1[tid] = si; s2[tid] = mr; s3[tid] = mi;
  __syncthreads();
  for (int s = 128; s > 0; s >>= 1) {
    if (tid < s) {
      s0[tid] += s0[tid + s];
      s1[tid] += s1[tid + s];
      s2[tid] = fmaxf(s2[tid], s2[tid + s]);
      s3[tid] = fmaxf(s3[tid], s3[tid + s]);
    }
    __syncthreads();
  }
  if (tid == 0) {
    ws[ch]         = s0[0] * (1.0f / HW_);
    ws[4096 + ch]  = s1[0] * (1.0f / HW_);
    ws[8192 + ch]  = s2[0];
    ws[12288 + ch] = s3[0];
  }
}

// ---------------------------------------------------------------------------
// Kernel 2: shared complex MLP on pooled features + sigmoid gates, via
// V_WMMA_F32_16X16X4_F32 (exact fp32, M=16 == batch). Single 8-wave block.
// ---------------------------------------------------------------------------
__device__ __forceinline__ v8f wmma4(v2f a, v2f b, v8f c) {
  // (neg_a, A, neg_b, B, c_mod, C, reuse_a, reuse_b); f32 NEG must be 0.
  return __builtin_amdgcn_wmma_f32_16x16x4_f32(false, a, false, b, (short)0, c,
                                               false, false);
}

__global__ __launch_bounds__(256) void gate_kernel(
    const float* __restrict__ w1r, const float* __restrict__ w1i,
    const float* __restrict__ b1r, const float* __restrict__ b1i,
    const float* __restrict__ w2r, const float* __restrict__ w2i,
    const float* __restrict__ b2r, const float* __restrict__ b2i,
    float* __restrict__ ws) {
  __shared__ float lds_h[2 * 2 * B_ * R_];  // [path][r/i][m=16][n=64]

  const int tid  = threadIdx.x;
  const int wave = tid >> 5;   // 0..7 (wave32)
  const int lane = tid & 31;
  const int half = lane >> 4;  // selects K-pair {k,k+1} vs {k+2,k+3}
  const int l16  = lane & 15;

  // ---- Phase 1: H = cleaky(clinear(Z, W1^T, b1)), Z = pooled [16,256] ----
  {
    const int p  = wave >> 2;              // 0 = avg path, 1 = max path
    const int nt = wave & 3;               // hidden n-tile (R=64 -> 4 tiles)
    const float* zr = ws + p * 8192;       // avg_r / max_r
    const float* zi = zr + 4096;           // avg_i / max_i
    const int n = nt * 16 + l16;           // hidden unit index 0..63

    v8f c_rr = {}, c_ii = {}, c_ri = {}, c_ir = {};
    for (int k0 = 0; k0 < C_; k0 += 4) {
      const int ka = k0 + 2 * half;
      v2f ar, ai, br, bi;
      ar.x = zr[l16 * C_ + ka];      ar.y = zr[l16 * C_ + ka + 1];
      ai.x = zi[l16 * C_ + ka];      ai.y = zi[l16 * C_ + ka + 1];
      br.x = w1r[n * C_ + ka];       br.y = w1r[n * C_ + ka + 1];   // B[k][n]=W1[n][k]
      bi.x = w1i[n * C_ + ka];       bi.y = w1i[n * C_ + ka + 1];
      c_rr = wmma4(ar, br, c_rr);    // Zr @ W1r^T
      c_ii = wmma4(ai, bi, c_ii);    // Zi @ W1i^T
      c_ri = wmma4(ar, bi, c_ri);    // Zr @ W1i^T
      c_ir = wmma4(ai, br, c_ir);    // Zi @ W1r^T
    }
    const float biasr = b1r[n], biasi = b1i[n];
#pragma unroll
    for (int j = 0; j < 8; ++j) {
      const int m = j + 8 * half;    // C/D layout: VGPR j -> rows j / j+8
      float hr = c_rr[j] - c_ii[j] + biasr;
      float hi = c_ri[j] + c_ir[j] + biasi;
      hr = hr > 0.f ? hr : 0.01f * hr;   // complex LeakyReLU
      hi = hi > 0.f ? hi : 0.01f * hi;
      lds_h[((p * 2 + 0) * B_ + m) * R_ + n] = hr;
      lds_h[((p * 2 + 1) * B_ + m) * R_ + n] = hi;
    }
  }
  __syncthreads();

  // ---- Phase 2: gate = sigmoid((avg + max) layer-2 outputs) ----
  // 32 jobs: {real,imag} x 16 n-tiles of C=256. Wave-uniform job loop.
  for (int job = wave; job < 32; job += 8) {
    const int ri = job >> 4;               // 0 real gate, 1 imag gate
    const int nt = job & 15;
    const int n  = nt * 16 + l16;          // output channel 0..255

    v8f acc = {};
    for (int k0 = 0; k0 < R_; k0 += 4) {
      const int ka = k0 + 2 * half;
      const float br0 = w2r[n * R_ + ka], br1 = w2r[n * R_ + ka + 1];
      const float bi0 = w2i[n * R_ + ka], bi1 = w2i[n * R_ + ka + 1];
      v2f bx, by;  // real: Hr*W2r - Hi*W2i ; imag: Hr*W2i + Hi*W2r
      if (ri == 0) { bx.x = br0; bx.y = br1; by.x = -bi0; by.y = -bi1; }
      else         { bx.x = bi0; bx.y = bi1; by.x =  br0; by.y =  br1; }
#pragma unroll
      for (int p = 0; p < 2; ++p) {        // accumulate avg + max paths
        v2f ahr, ahi;
        ahr.x = lds_h[((p * 2 + 0) * B_ + l16) * R_ + ka];
        ahr.y = lds_h[((p * 2 + 0) * B_ + l16) * R_ + ka + 1];
        ahi.x = lds_h[((p * 2 + 1) * B_ + l16) * R_ + ka];
        ahi.y = lds_h[((p * 2 + 1) * B_ + l16) * R_ + ka + 1];
        acc = wmma4(ahr, bx, acc);
        acc = wmma4(ahi, by, acc);
      }
    }
    const float bias2 = 2.0f * (ri == 0 ? b2r[n] : b2i[n]);  // bias in a and m
#pragma unroll
    for (int j = 0; j < 8; ++j) {
      const int m = j + 8 * half;
      const float v = acc[j] + bias2;
      const float g = 1.0f / (1.0f + expf(-v));
      ws[16384 + ri * 4096 + m * C_ + n] = g;
    }
  }
}

// ---------------------------------------------------------------------------
// Kernel 3: complex gating, pure streaming. float4 per thread, NT hints.
// out_r = xr*gr - xi*gi ; out_i = xr*gi + xi*gr ; out = [2,B,C,H,W]
// ---------------------------------------------------------------------------
__global__ __launch_bounds__(256) void apply_kernel(const float* __restrict__ xr,
                                                    const float* __restrict__ xi,
                                                    const float* __restrict__ ws,
                                                    float* __restrict__ out) {
  const long long idx = (long long)blockIdx.x * 256 + threadIdx.x;  // float4 idx
  const int ch = (int)(idx >> 12);          // 4096 float4 per channel (uniform/wave)
  const float gr = ws[16384 + ch];
  const float gi = ws[20480 + ch];

  const v4f* xr4 = (const v4f*)xr;
  const v4f* xi4 = (const v4f*)xi;
  v4f a = __builtin_nontemporal_load(&xr4[idx]);
  v4f b = __builtin_nontemporal_load(&xi4[idx]);
  v4f o_r = a * gr - b * gi;
  v4f o_i = a * gi + b * gr;

  v4f* out_r = (v4f*)out;
  v4f* out_i = out_r + ((long long)NCH * (HW_ / 4));  // second plane
  __builtin_nontemporal_store(o_r, &out_r[idx]);
  __builtin_nontemporal_store(o_i, &out_i[idx]);
}

// ---------------------------------------------------------------------------
extern "C" void kernel_launch(void* const* d_in, const int* in_sizes, int n_in,
                              void* d_out, int out_size, void* d_ws, size_t ws_size,
                              hipStream_t stream) {
  const float* xr  = (const float*)d_in[0];
  const float* xi  = (const float*)d_in[1];
  const float* w1r = (const float*)d_in[2];
  const float* w1i = (const float*)d_in[3];
  const float* b1r = (const float*)d_in[4];
  const float* b1i = (const float*)d_in[5];
  const float* w2r = (const float*)d_in[6];
  const float* w2i = (const float*)d_in[7];
  const float* b2r = (const float*)d_in[8];
  const float* b2i = (const float*)d_in[9];
  float* ws  = (float*)d_ws;
  float* out = (float*)d_out;

  pool_kernel<<<NCH, 256, 0, stream>>>(xr, xi, ws);
  gate_kernel<<<1, 256, 0, stream>>>(w1r, w1i, b1r, b1i, w2r, w2i, b2r, b2i, ws);
  apply_kernel<<<(NCH * (HW_ / 4)) / 256, 256, 0, stream>>>(xr, xi, ws, out);
}